// EdgePredictor_55267639165304
// MI455X (gfx1250) — compile-verified
//
#include <hip/hip_runtime.h>

typedef __attribute__((ext_vector_type(2))) float v2f;
typedef __attribute__((ext_vector_type(8))) float v8f;

#define DD 128   // feature dim

// ---------------------------------------------------------------------------
// Kernel 1: per-node projections via fp32 WMMA.
//   U = z @ W1[0:128, :]   (which == 0)
//   V = z @ W1[128:256, :] (which == 1)
// One wave computes a 16(M) x 128(N) output tile of one half.
// A-frag layout (V_WMMA_F32_16X16X4_F32): lanes 0-15 -> M=0..15, VGPR0/1 = K,K+1;
// lanes 16-31 -> same M rows, K+2,K+3.
// ---------------------------------------------------------------------------
__global__ __launch_bounds__(256) void ep_proj_kernel(
    const float* __restrict__ z, const float* __restrict__ W1,
    float* __restrict__ U, float* __restrict__ V, int nMTiles, int N)
{
    const int wave  = (blockIdx.x * blockDim.x + threadIdx.x) >> 5;
    const int lane  = threadIdx.x & 31;
    const int which = wave & 1;
    const int mtile = wave >> 1;
    if (mtile >= nMTiles) return;          // whole-wave exit -> EXEC stays all-ones

    const float* __restrict__ B  = W1 + (size_t)which * DD * DD; // [128 x 128] row-major
    float*       __restrict__ Oo = which ? V : U;

    const int half  = lane >> 4;           // 0: K,K+1   1: K+2,K+3
    const int lr    = lane & 15;
    const int mbase = mtile * 16;
    const int arow  = min(mbase + lr, N - 1);   // clamp tail reads
    const float* __restrict__ zrow = z + (size_t)arow * DD;

    v8f acc[8];
#pragma unroll
    for (int i = 0; i < 8; ++i) { v8f zf = {}; acc[i] = zf; }

    for (int k = 0; k < DD; k += 4) {
        const int kk = k + 2 * half;
        v2f a;
        a.x = zrow[kk];
        a.y = zrow[kk + 1];
#pragma unroll
        for (int nt = 0; nt < 8; ++nt) {
            v2f b;                                    // B[k][n] frag: VGPRv = K=kk+v
            b.x = B[(size_t)kk       * DD + nt * 16 + lr];
            b.y = B[(size_t)(kk + 1) * DD + nt * 16 + lr];
            acc[nt] = __builtin_amdgcn_wmma_f32_16x16x4_f32(
                false, a, false, b, (short)0, acc[nt], false, false);
        }
    }

    // C layout: VGPR v, lanes 0-15 -> M=v, N=lane; lanes 16-31 -> M=v+8, N=lane-16
#pragma unroll
    for (int nt = 0; nt < 8; ++nt) {
#pragma unroll
        for (int v = 0; v < 8; ++v) {
            const int row = mbase + v + 8 * half;
            if (row < N)
                Oo[(size_t)row * DD + nt * 16 + lr] = acc[nt][v];
        }
    }
}

// ---------------------------------------------------------------------------
// Kernel 2: per-edge fused gather + bias + relu + dot(W2) + b2.
// One wave32 per edge; lane handles a float4 chunk of the 128-wide row.
// U, V are L2-resident (51 MB each on a 192 MB L2).
// ---------------------------------------------------------------------------
__global__ __launch_bounds__(256) void ep_edge_kernel(
    const float* __restrict__ U, const float* __restrict__ V,
    const long long* __restrict__ ei,     // [2, E] int64
    const float* __restrict__ b1, const float* __restrict__ W2,
    const float* __restrict__ b2, float* __restrict__ out, int E)
{
    const int wave = (blockIdx.x * blockDim.x + threadIdx.x) >> 5;
    const int lane = threadIdx.x & 31;
    if (wave >= E) return;

    const long long r = ei[wave];
    const long long c = ei[(size_t)E + wave];

    const float4 u  = ((const float4*)(U + (size_t)r * DD))[lane];
    const float4 v  = ((const float4*)(V + (size_t)c * DD))[lane];
    const float4 bb = ((const float4*)b1)[lane];
    const float4 w  = ((const float4*)W2)[lane];

    float p = fmaxf(u.x + v.x + bb.x, 0.f) * w.x;
    p = fmaf(fmaxf(u.y + v.y + bb.y, 0.f), w.y, p);
    p = fmaf(fmaxf(u.z + v.z + bb.z, 0.f), w.z, p);
    p = fmaf(fmaxf(u.w + v.w + bb.w, 0.f), w.w, p);

#pragma unroll
    for (int off = 16; off > 0; off >>= 1)
        p += __shfl_down(p, off, 32);

    if (lane == 0) out[wave] = p + b2[0];
}

// ---------------------------------------------------------------------------
extern "C" void kernel_launch(void* const* d_in, const int* in_sizes, int n_in,
                              void* d_out, int out_size, void* d_ws, size_t ws_size,
                              hipStream_t stream)
{
    const float*     z  = (const float*)d_in[0];      // [N, 128]
    const long long* ei = (const long long*)d_in[1];  // [2, E] int64
    const float*     W1 = (const float*)d_in[2];      // [256, 128]
    const float*     b1 = (const float*)d_in[3];      // [128]
    const float*     W2 = (const float*)d_in[4];      // [128]
    const float*     b2 = (const float*)d_in[5];      // [1]
    float*           out = (float*)d_out;

    const int N = in_sizes[0] / DD;
    const int E = in_sizes[1] / 2;

    float* U = (float*)d_ws;                 // [N, 128]
    float* V = U + (size_t)N * DD;           // [N, 128]

    const int nMTiles  = (N + 15) / 16;
    const int projWaves = nMTiles * 2;                         // (mtile, which)
    const int projBlocks = (projWaves * 32 + 255) / 256;
    ep_proj_kernel<<<projBlocks, 256, 0, stream>>>(z, W1, U, V, nMTiles, N);

    const int edgeBlocks = (E + 7) / 8;                        // 8 waves/block
    ep_edge_kernel<<<edgeBlocks, 256, 0, stream>>>(U, V, ei, b1, W2, b2, out, E);
}